// GraphSAGE_28303834480830
// MI455X (gfx1250) — compile-verified
//
#include <hip/hip_runtime.h>
#include <hip/hip_bf16.h>

typedef float v2f __attribute__((ext_vector_type(2)));
typedef float v8f __attribute__((ext_vector_type(8)));

// ---------------------------------------------------------------------------
// Problem constants (fixed by the reference)
// ---------------------------------------------------------------------------
#define N_S 100000
#define N_C 20000
#define N_L 5000
#define D_S 64
#define D_C 32
#define D_L 128
#define H   128

// ---------------------------------------------------------------------------
// GEMM parameters (passed by value). out[M x 128] = act(scale*(sum_c A_c@W_c + bias))
// W matrices are K x 128 row-major. Wadd[c] (optional) is added to W[c] on LDS fill.
// ---------------------------------------------------------------------------
struct GemmParams {
  const float* A[3];
  const float* W[3];
  const float* Wadd[3];
  const float* bias0;
  const float* bias1;
  float scale;
  int relu;
  int M;
  float* out;
};

// Block: 128 threads = 4 waves. Each block computes a 64x128 output tile,
// each wave a 16x128 strip (8 WMMA accumulator tiles of 16x16 f32).
//
// LDS layouts:
//  A: row-major [64][K], row stride K+4 words -> 16 row-strided b64 reads per
//     half-wave hit distinct banks (stride (K+4)%64 == 4, halves offset by 2).
//  B: k-pair interleaved [K/2][128 pairs], pair-row stride 288 words
//     (288%64==32 -> the two lane-halves, which read consecutive k-pairs, use
//     disjoint bank sets). Each WMMA B fragment (rows k,k+1 at one column) is
//     then a single aligned ds_load_b64 into an even VGPR pair -> no repack movs.
template <int K, int NCHUNK>
__global__ __launch_bounds__(128) void gemm_wmma_f32(GemmParams p) {
  constexpr int ASTRIDE = K + 4;  // words per A row
  constexpr int BSTRIDE = 288;    // words per B k-pair row (128*2 + 32 pad)
  __shared__ float lds_a[64 * ASTRIDE];
  __shared__ float lds_b[(K / 2) * BSTRIDE];

  const int tid   = threadIdx.x;
  const int wave  = tid >> 5;
  const int lane  = tid & 31;
  const int lhalf = lane >> 4;   // 0: lanes 0-15, 1: lanes 16-31
  const int lmod  = lane & 15;
  const int row0  = blockIdx.x * 64;  // global row base of this block
  const int wrow  = wave * 16;        // wave's row offset within the tile

  v8f acc[8];
#pragma unroll
  for (int t = 0; t < 8; ++t) acc[t] = {};

  for (int c = 0; c < NCHUNK; ++c) {
    const float* A    = p.A[c];
    const float* W    = p.W[c];
    const float* Wadd = p.Wadd[c];
    __syncthreads();  // previous chunk's LDS reads complete
    // ---- stage W (+Wadd) : K x 128, coalesced, k-pair interleaved ----
    for (int i = tid; i < K * 128; i += 128) {
      int r = i >> 7, col = i & 127;
      float w = W[i];
      if (Wadd) w += Wadd[i];
      lds_b[(r >> 1) * BSTRIDE + col * 2 + (r & 1)] = w;
    }
    // ---- stage A tile : 64 x K, coalesced, zero-pad past M ----
    for (int i = tid; i < 64 * K; i += 128) {
      int r = i / K, col = i % K;  // K is a power-of-two template constant
      int gr = row0 + r;
      lds_a[r * ASTRIDE + col] = (gr < p.M) ? A[gr * K + col] : 0.0f;
    }
    __syncthreads();

    // ---- WMMA main loop: K/4 steps of 16x16x4 f32 ----
    for (int k0 = 0; k0 < K; k0 += 4) {
      // A fragment (ISA layout): lane m=lmod holds K = k0+2*lhalf, +1
      v2f a = *(const v2f*)&lds_a[(wrow + lmod) * ASTRIDE + k0 + 2 * lhalf];
      const int kp = (k0 >> 1) + lhalf;  // k-pair index for this lane half
#pragma unroll
      for (int t = 0; t < 8; ++t) {
        // single b64 load: rows (k0+2*lhalf, +1) at column t*16+lmod
        v2f b = *(const v2f*)&lds_b[kp * BSTRIDE + (t * 16 + lmod) * 2];
        acc[t] = __builtin_amdgcn_wmma_f32_16x16x4_f32(
            false, a, false, b, (short)0, acc[t], false, false);
      }
    }
  }

  // ---- epilogue: bias, scale, relu, store (C/D layout: VGPR r -> M=r+8*lhalf) ----
#pragma unroll
  for (int t = 0; t < 8; ++t) {
    const int n = t * 16 + lmod;
    float bv = 0.0f;
    if (p.bias0) bv += p.bias0[n];
    if (p.bias1) bv += p.bias1[n];
#pragma unroll
    for (int r = 0; r < 8; ++r) {
      int m = row0 + wrow + r + 8 * lhalf;
      if (m < p.M) {
        float v = (acc[t][r] + bv) * p.scale;
        if (p.relu) v = fmaxf(v, 0.0f);
        p.out[m * 128 + n] = v;
      }
    }
  }
}

// ---------------------------------------------------------------------------
// Edge scatter: one wave per edge, lane l handles float4 channels [4l,4l+4).
// nsum[d] += h[s], deg[d] += 1. Coalesced 512B gather + 4 f32 atomics per lane.
// ---------------------------------------------------------------------------
__global__ __launch_bounds__(256) void scatter_kernel(
    const float* __restrict__ h, const int* __restrict__ src,
    const int* __restrict__ dst, float* __restrict__ nsum,
    float* __restrict__ deg, int nedges) {
  const int gwave  = (blockIdx.x * blockDim.x + threadIdx.x) >> 5;
  const int lane   = threadIdx.x & 31;
  const int nwaves = (gridDim.x * blockDim.x) >> 5;
  for (int e = gwave; e < nedges; e += nwaves) {
    const int s = src[e];
    const int d = dst[e];
    const float4 v = ((const float4*)(h + (size_t)s * 128))[lane];
    float* o = nsum + (size_t)d * 128 + lane * 4;
    atomicAdd(o + 0, v.x);
    atomicAdd(o + 1, v.y);
    atomicAdd(o + 2, v.z);
    atomicAdd(o + 3, v.w);
    if (lane == 0) atomicAdd(deg + d, 1.0f);
  }
}

__global__ void zero_kernel(float* __restrict__ p, int n) {
  int i = blockIdx.x * blockDim.x + threadIdx.x;
  if (i < n) p[i] = 0.0f;
}

// nsum[r][c] /= max(deg[r], 1)
__global__ void divide_kernel(float* __restrict__ nsum,
                              const float* __restrict__ deg, int nelem) {
  int i = blockIdx.x * blockDim.x + threadIdx.x;
  if (i < nelem) nsum[i] /= fmaxf(deg[i >> 7], 1.0f);
}

// ---------------------------------------------------------------------------
// Host launcher
// ---------------------------------------------------------------------------
static GemmParams mk1(const float* A, const float* W, const float* b, int M,
                      float* out) {
  GemmParams p{};
  p.A[0] = A; p.W[0] = W; p.Wadd[0] = nullptr;
  p.bias0 = b; p.bias1 = nullptr;
  p.scale = 1.0f; p.relu = 0; p.M = M; p.out = out;
  return p;
}

extern "C" void kernel_launch(void* const* d_in, const int* in_sizes, int n_in,
                              void* d_out, int out_size, void* d_ws,
                              size_t ws_size, hipStream_t stream) {
  const float* feat_s = (const float*)d_in[0];
  const float* feat_c = (const float*)d_in[1];
  const float* feat_l = (const float*)d_in[2];
  const float* W_fs = (const float*)d_in[3];
  const float* b_fs = (const float*)d_in[4];
  const float* W_fc = (const float*)d_in[5];
  const float* b_fc = (const float*)d_in[6];
  const float* W_fl = (const float*)d_in[7];
  const float* b_fl = (const float*)d_in[8];
  const float* W_self_u  = (const float*)d_in[9];
  const float* W_neigh_u = (const float*)d_in[10];
  const float* b_u = (const float*)d_in[11];
  const float* W_self_t  = (const float*)d_in[12];
  const float* W_neigh_t = (const float*)d_in[13];
  const float* b_t = (const float*)d_in[14];
  const int* und_src = (const int*)d_in[15];
  const int* und_dst = (const int*)d_in[16];
  const int* tea_src = (const int*)d_in[17];
  const int* tea_dst = (const int*)d_in[18];
  const int E_U = in_sizes[15];
  const int E_T = in_sizes[17];
  float* out = (float*)d_out;

  // workspace layout (floats)
  float* ws    = (float*)d_ws;
  float* h_s   = ws;                       // N_S*128
  float* h_l   = h_s + (size_t)N_S * H;    // N_L*128
  float* hnu   = h_l + (size_t)N_L * H;    // N_C*128 (sum -> mean, student relation)
  float* hnt   = hnu + (size_t)N_C * H;    // N_C*128 (lecture relation)
  float* deg_u = hnt + (size_t)N_C * H;    // N_C
  float* deg_t = deg_u + N_C;              // N_C
  float* hcA   = deg_t + N_C;              // N_C*128
  float* hcB   = hcA + (size_t)N_C * H;    // N_C*128

  // 1) zero accumulators (hnu, hnt, deg_u, deg_t are contiguous)
  {
    int nz = 2 * N_C * H + 2 * N_C;
    zero_kernel<<<(nz + 255) / 256, 256, 0, stream>>>(hnu, nz);
  }

  // 2) input projections  h = feat @ W + b
  gemm_wmma_f32<D_S, 1><<<(N_S + 63) / 64, 128, 0, stream>>>(
      mk1(feat_s, W_fs, b_fs, N_S, h_s));
  gemm_wmma_f32<D_L, 1><<<(N_L + 63) / 64, 128, 0, stream>>>(
      mk1(feat_l, W_fl, b_fl, N_L, h_l));
  gemm_wmma_f32<D_C, 1><<<(N_C + 63) / 64, 128, 0, stream>>>(
      mk1(feat_c, W_fc, b_fc, N_C, hcA));

  // 3) layer-invariant neighbor means (src node types never change)
  scatter_kernel<<<4096, 256, 0, stream>>>(h_s, und_src, und_dst, hnu, deg_u, E_U);
  scatter_kernel<<<2048, 256, 0, stream>>>(h_l, tea_src, tea_dst, hnt, deg_t, E_T);
  divide_kernel<<<(N_C * H + 255) / 256, 256, 0, stream>>>(hnu, deg_u, N_C * H);
  divide_kernel<<<(N_C * H + 255) / 256, 256, 0, stream>>>(hnt, deg_t, N_C * H);

  // 4) three fused hetero-SAGE layers on concept nodes
  const float* hc_in[3]  = {hcA, hcB, hcA};
  float*       hc_out[3] = {hcB, hcA, out};
  for (int i = 0; i < 3; ++i) {
    GemmParams p{};
    p.A[0] = hc_in[i];  p.W[0] = W_self_u  + (size_t)i * H * H;
    p.Wadd[0] = W_self_t + (size_t)i * H * H;   // fuse the two self-GEMMs
    p.A[1] = hnu;       p.W[1] = W_neigh_u + (size_t)i * H * H; p.Wadd[1] = nullptr;
    p.A[2] = hnt;       p.W[2] = W_neigh_t + (size_t)i * H * H; p.Wadd[2] = nullptr;
    p.bias0 = b_u + (size_t)i * H;
    p.bias1 = b_t + (size_t)i * H;
    p.scale = (i == 0) ? 0.5f : 1.0f;   // HeteroGraphConv 'mean' on layer 0
    p.relu  = (i != 2) ? 1 : 0;
    p.M     = N_C;
    p.out   = hc_out[i];
    gemm_wmma_f32<H, 3><<<(N_C + 63) / 64, 128, 0, stream>>>(p);
  }
}